// RegulatoryGNN_83382495085223
// MI455X (gfx1250) — compile-verified
//
#include <hip/hip_runtime.h>
#include <hip/hip_bf16.h>
#include <cmath>

// ---------------------------------------------------------------------------
// RegulatoryGNN: 2-layer GAT + mean-pool + sigmoid classifier for gfx1250.
// Dense GEMMs via V_WMMA_F32_16X16X4_F32 (full fp32 precision).
// ---------------------------------------------------------------------------

typedef __attribute__((ext_vector_type(2))) float v2f;
typedef __attribute__((ext_vector_type(8))) float v8f;

#define N_NODES  50000
#define N_EDGES  800000
#define E_TOT    (N_EDGES + N_NODES)   // edges + self loops
#define IN_DIM   256
#define HID      128
#define N_GRAPHS 64
#define NEG_SLOPE 0.2f

// ---------------------------------------------------------------------------
// GEMM: C[M,128] = A[M,K] @ B[K,128], optional ReLU applied to A on load.
// One block = 16 rows of A; 8 waves, each owns one 16-col tile of the output.
// A tile staged in LDS with +4 padding (stride 260 -> bank-conflict free).
// ---------------------------------------------------------------------------
__global__ __launch_bounds__(256) void gat_gemm_wmma(
    const float* __restrict__ A, const float* __restrict__ B,
    float* __restrict__ C, int M, int K, int relu_in)
{
    __shared__ float As[16 * 260];           // supports K up to 256
    const int m0  = blockIdx.x * 16;
    const int tid = threadIdx.x;

    // Stage 16 x K tile of A (coalesced), fuse ReLU of previous layer.
    for (int i = tid; i < 16 * K; i += 256) {
        int r = i / K, c = i - r * K;
        float v = A[(size_t)(m0 + r) * K + c];
        if (relu_in) v = fmaxf(v, 0.0f);
        As[r * 260 + c] = v;
    }
    __syncthreads();

    const int wave = tid >> 5;               // 0..7 -> N tile
    const int lane = tid & 31;
    const int l16  = lane & 15;
    const int half = lane >> 4;              // 0: K+0/K+1, 1: K+2/K+3
    const int n0   = wave * 16;
    const int n    = n0 + l16;

    v8f acc = {};
    const float* Arow = &As[l16 * 260];
    for (int k = 0; k < K; k += 4) {
        v2f a, b;
        a.x = Arow[k + 2 * half + 0];
        a.y = Arow[k + 2 * half + 1];
        const float* Bp = B + (size_t)(k + 2 * half) * HID + n;
        b.x = Bp[0];
        b.y = Bp[HID];
        // D = A(16x4,f32) * B(4x16,f32) + C(16x16,f32)
        acc = __builtin_amdgcn_wmma_f32_16x16x4_f32(
            false, a, false, b, (short)0, acc, false, false);
    }

    // D layout: VGPR i -> M = m0 + half*8 + i, N = n (lanes 0-15 / 16-31)
    const int mbase = m0 + half * 8;
#pragma unroll
    for (int i = 0; i < 8; ++i)
        C[(size_t)(mbase + i) * HID + n] = acc[i];
}

// ---------------------------------------------------------------------------
// Per-node attention logits: alpha_s[i] = h[i,:]·a_src, alpha_d[i] = h[i,:]·a_dst
// One wave32 per node, shuffle reduction.
// ---------------------------------------------------------------------------
__global__ __launch_bounds__(256) void gat_alphas(
    const float* __restrict__ h,
    const float* __restrict__ a_s, const float* __restrict__ a_d,
    float* __restrict__ alpha_s, float* __restrict__ alpha_d, int M)
{
    int node = (blockIdx.x * 256 + threadIdx.x) >> 5;
    int lane = threadIdx.x & 31;
    if (node >= M) return;
    const float* hp = h + (size_t)node * HID;
    float s = 0.f, d = 0.f;
#pragma unroll
    for (int f = lane; f < HID; f += 32) {
        float v = hp[f];
        s += v * a_s[f];
        d += v * a_d[f];
    }
    for (int off = 16; off; off >>= 1) {
        s += __shfl_down(s, off);
        d += __shfl_down(d, off);
    }
    if (lane == 0) { alpha_s[node] = s; alpha_d[node] = d; }
}

// ---------------------------------------------------------------------------
__device__ inline float atomicMaxFloat(float* addr, float val)
{
    // works for all finite floats via ordered-int trick
    if (val >= 0.0f)
        return __int_as_float(atomicMax((int*)addr, __float_as_int(val)));
    else
        return __uint_as_float(atomicMin((unsigned int*)addr, __float_as_uint(val)));
}

__global__ __launch_bounds__(256) void gat_init_node(
    float* __restrict__ m, float* __restrict__ denom, int N)
{
    int i = blockIdx.x * 256 + threadIdx.x;
    if (i < N) { m[i] = -INFINITY; denom[i] = 0.0f; }
}

__global__ __launch_bounds__(256) void gat_init_bias(
    const float* __restrict__ b, float* __restrict__ out, int N)
{
    int i = blockIdx.x * 256 + threadIdx.x;
    if (i < N * HID) out[i] = b[i & (HID - 1)];
}

// e = leaky_relu(alpha_s[src] + alpha_d[dst]); segment-max into m[dst]
__global__ __launch_bounds__(256) void gat_edge_max(
    const int* __restrict__ src, const int* __restrict__ dst,
    const float* __restrict__ alpha_s, const float* __restrict__ alpha_d,
    float* __restrict__ ew, float* __restrict__ m)
{
    int i = blockIdx.x * 256 + threadIdx.x;
    if (i >= E_TOT) return;
    int s, d;
    if (i < N_EDGES) { s = src[i]; d = dst[i]; }
    else             { s = d = i - N_EDGES; }          // self loop
    float e = alpha_s[s] + alpha_d[d];
    e = (e > 0.0f) ? e : NEG_SLOPE * e;
    ew[i] = e;
    atomicMaxFloat(&m[d], e);
}

// w = exp(e - m[dst]); segment-sum into denom[dst]
__global__ __launch_bounds__(256) void gat_edge_expsum(
    const int* __restrict__ dst, const float* __restrict__ m,
    float* __restrict__ ew, float* __restrict__ denom)
{
    int i = blockIdx.x * 256 + threadIdx.x;
    if (i >= E_TOT) return;
    int d = (i < N_EDGES) ? dst[i] : (i - N_EDGES);
    float w = expf(ew[i] - m[d]);
    ew[i] = w;
    atomicAdd(&denom[d], w);
}

// out[dst,:] += h[src,:] * (w / denom[dst]) ; one wave32 per edge
__global__ __launch_bounds__(256) void gat_edge_scatter(
    const int* __restrict__ src, const int* __restrict__ dst,
    const float* __restrict__ ew, const float* __restrict__ denom,
    const float* __restrict__ h, float* __restrict__ out)
{
    int e    = (blockIdx.x * 256 + threadIdx.x) >> 5;
    int lane = threadIdx.x & 31;
    if (e >= E_TOT) return;
    int s, d;
    if (e < N_EDGES) { s = src[e]; d = dst[e]; }
    else             { s = d = e - N_EDGES; }
    float coef = ew[e] / denom[d];
    const float* hp = h + (size_t)s * HID;
    float* op = out + (size_t)d * HID;
#pragma unroll
    for (int f = lane; f < HID; f += 32)
        atomicAdd(&op[f], hp[f] * coef);
}

// ---------------------------------------------------------------------------
__global__ __launch_bounds__(256) void gat_init_pool(
    float* __restrict__ pooled, int* __restrict__ counts)
{
    int i = blockIdx.x * 256 + threadIdx.x;
    if (i < N_GRAPHS * HID) pooled[i] = 0.0f;
    if (i < N_GRAPHS)       counts[i] = 0;
}

// pooled[batch[n],:] += relu(agg2[n,:]) ; counts[batch[n]] += 1
__global__ __launch_bounds__(256) void gat_pool(
    const float* __restrict__ agg, const int* __restrict__ batch,
    float* __restrict__ pooled, int* __restrict__ counts)
{
    int i = blockIdx.x * 256 + threadIdx.x;
    if (i >= N_NODES * HID) return;
    int node = i >> 7, f = i & (HID - 1);
    int g = batch[node];
    atomicAdd(&pooled[g * HID + f], fmaxf(agg[i], 0.0f));
    if (f == 0) atomicAdd(&counts[g], 1);
}

// out[g] = sigmoid( (pooled[g,:]/max(cnt,1)) · Wc + bc ) ; one wave per graph
__global__ __launch_bounds__(32) void gat_classify(
    const float* __restrict__ pooled, const int* __restrict__ counts,
    const float* __restrict__ Wc, const float* __restrict__ bc,
    float* __restrict__ out)
{
    int g = blockIdx.x;
    int lane = threadIdx.x;
    float cnt = fmaxf((float)counts[g], 1.0f);
    float sum = 0.f;
#pragma unroll
    for (int f = lane; f < HID; f += 32)
        sum += (pooled[g * HID + f] / cnt) * Wc[f];
    for (int off = 16; off; off >>= 1) sum += __shfl_down(sum, off);
    if (lane == 0) out[g] = 1.0f / (1.0f + expf(-(sum + bc[0])));
}

// ---------------------------------------------------------------------------
extern "C" void kernel_launch(void* const* d_in, const int* in_sizes, int n_in,
                              void* d_out, int out_size, void* d_ws, size_t ws_size,
                              hipStream_t stream)
{
    const float* x       = (const float*)d_in[0];
    const int*   eidx    = (const int*)  d_in[1];   // [2, 800000]
    const int*   batch   = (const int*)  d_in[2];
    const float* W1      = (const float*)d_in[3];
    const float* a_src1  = (const float*)d_in[4];
    const float* a_dst1  = (const float*)d_in[5];
    const float* b1      = (const float*)d_in[6];
    const float* W2      = (const float*)d_in[7];
    const float* a_src2  = (const float*)d_in[8];
    const float* a_dst2  = (const float*)d_in[9];
    const float* b2      = (const float*)d_in[10];
    const float* Wc      = (const float*)d_in[11];
    const float* bc      = (const float*)d_in[12];
    float* out = (float*)d_out;

    const int* src = eidx;
    const int* dst = eidx + N_EDGES;

    // ---- workspace carve-out (256B aligned) ----
    char* p = (char*)d_ws;
    auto carve = [&](size_t bytes) {
        char* r = p;
        p += (bytes + 255) & ~(size_t)255;
        return r;
    };
    float* h1      = (float*)carve((size_t)N_NODES * HID * 4);
    float* agg1    = (float*)carve((size_t)N_NODES * HID * 4);
    float* h2      = (float*)carve((size_t)N_NODES * HID * 4);
    float* agg2    = (float*)carve((size_t)N_NODES * HID * 4);
    float* alpha_s = (float*)carve((size_t)N_NODES * 4);
    float* alpha_d = (float*)carve((size_t)N_NODES * 4);
    float* segmax  = (float*)carve((size_t)N_NODES * 4);
    float* denom   = (float*)carve((size_t)N_NODES * 4);
    float* ew      = (float*)carve((size_t)E_TOT * 4);
    float* pooled  = (float*)carve((size_t)N_GRAPHS * HID * 4);
    int*   counts  = (int*)  carve((size_t)N_GRAPHS * 4);

    const int gemm_blocks   = N_NODES / 16;                       // 3125, exact
    const int alpha_blocks  = (N_NODES * 32 + 255) / 256;
    const int node_blocks   = (N_NODES + 255) / 256;
    const int nodeF_blocks  = (N_NODES * HID + 255) / 256;
    const int edge_blocks   = (E_TOT + 255) / 256;
    const int scat_blocks   = ((size_t)E_TOT * 32 + 255) / 256;

    // ================= layer 1 =================
    gat_gemm_wmma<<<gemm_blocks, 256, 0, stream>>>(x, W1, h1, N_NODES, IN_DIM, 0);
    gat_alphas<<<alpha_blocks, 256, 0, stream>>>(h1, a_src1, a_dst1, alpha_s, alpha_d, N_NODES);
    gat_init_node<<<node_blocks, 256, 0, stream>>>(segmax, denom, N_NODES);
    gat_init_bias<<<nodeF_blocks, 256, 0, stream>>>(b1, agg1, N_NODES);
    gat_edge_max<<<edge_blocks, 256, 0, stream>>>(src, dst, alpha_s, alpha_d, ew, segmax);
    gat_edge_expsum<<<edge_blocks, 256, 0, stream>>>(dst, segmax, ew, denom);
    gat_edge_scatter<<<scat_blocks, 256, 0, stream>>>(src, dst, ew, denom, h1, agg1);

    // ================= layer 2 (ReLU fused into GEMM A-load) =================
    gat_gemm_wmma<<<gemm_blocks, 256, 0, stream>>>(agg1, W2, h2, N_NODES, HID, 1);
    gat_alphas<<<alpha_blocks, 256, 0, stream>>>(h2, a_src2, a_dst2, alpha_s, alpha_d, N_NODES);
    gat_init_node<<<node_blocks, 256, 0, stream>>>(segmax, denom, N_NODES);
    gat_init_bias<<<nodeF_blocks, 256, 0, stream>>>(b2, agg2, N_NODES);
    gat_edge_max<<<edge_blocks, 256, 0, stream>>>(src, dst, alpha_s, alpha_d, ew, segmax);
    gat_edge_expsum<<<edge_blocks, 256, 0, stream>>>(dst, segmax, ew, denom);
    gat_edge_scatter<<<scat_blocks, 256, 0, stream>>>(src, dst, ew, denom, h2, agg2);

    // ================= pool + classify =================
    gat_init_pool<<<(N_GRAPHS * HID + 255) / 256, 256, 0, stream>>>(pooled, counts);
    gat_pool<<<nodeF_blocks, 256, 0, stream>>>(agg2, batch, pooled, counts);
    gat_classify<<<N_GRAPHS, 32, 0, stream>>>(pooled, counts, Wc, bc, out);

    (void)in_sizes; (void)n_in; (void)out_size; (void)ws_size;
}